// GraphConvolution_54726473285785
// MI455X (gfx1250) — compile-verified
//
#include <hip/hip_runtime.h>
#include <hip/hip_bf16.h>

typedef __attribute__((ext_vector_type(2))) float v2f;
typedef __attribute__((ext_vector_type(8))) float v8f;

#define F_IN  256
#define F_OUT 128

// -----------------------------------------------------------------------------
// Kernel 1: h = x @ w   (dense f32 GEMM via V_WMMA_F32_16X16X4_F32)
// Block = 128 threads = 4 waves; each wave owns a 16(M) x 128(N) output tile
// held in 8 v8f accumulators. K loop steps by 4 (WMMA K=4).
// A-operand layout (16x4 f32): lanes 0-15 -> M=lane, K={0,1} in the 2 VGPRs;
// lanes 16-31 -> M=lane-16, K={2,3}. One aligned float2 load per lane.
// B-operand layout (4x16 f32): lanes 0-15 -> N=lane, K={0,1}; lanes 16-31 ->
// N=lane-16, K={2,3}. Two b32 loads per lane (row stride F_OUT).
// C/D layout (16x16 f32): VGPR r, lanes 0-15 -> M=r, lanes 16-31 -> M=r+8.
// -----------------------------------------------------------------------------
__global__ __launch_bounds__(128) void gcn_gemm_wmma(
    const float* __restrict__ x, const float* __restrict__ w,
    float* __restrict__ h, int n_nodes)
{
    const int lane   = threadIdx.x & 31;
    const int waveId = threadIdx.x >> 5;
    const int m0     = (blockIdx.x * 4 + waveId) * 16;
    if (m0 + 16 > n_nodes) return;          // wave-uniform: EXEC stays all-1s

    const int row = lane & 15;              // M (for A) / N (for B/C) index
    const int kh  = lane >> 4;              // which K-pair this half-wave holds

    v8f acc[8];
#pragma unroll
    for (int t = 0; t < 8; ++t) acc[t] = (v8f)0.0f;

    const float* xrow = x + (size_t)(m0 + row) * F_IN + kh * 2;

    for (int k = 0; k < F_IN; k += 4) {
        v2f A = *(const v2f*)(xrow + k);    // 8B-aligned: k%4==0, kh*2 offset
#pragma unroll
        for (int t = 0; t < 8; ++t) {
            const float* bp = w + (size_t)(k + kh * 2) * F_OUT + t * 16 + row;
            v2f B;
            B.x = bp[0];
            B.y = bp[F_OUT];
            // 8 args: (neg_a, A, neg_b, B, c_mod, C, reuse_a, reuse_b)
            acc[t] = __builtin_amdgcn_wmma_f32_16x16x4_f32(
                false, A, false, B, (short)0, acc[t], false, false);
        }
    }

    // Writeback: acc[t][r] lives at row (m0 + r + kh*8), col (t*16 + row)
#pragma unroll
    for (int t = 0; t < 8; ++t) {
        float* hp = h + (size_t)(m0 + kh * 8) * F_OUT + t * 16 + row;
#pragma unroll
        for (int r = 0; r < 8; ++r)
            hp[(size_t)r * F_OUT] = acc[t][r];
    }
}

// -----------------------------------------------------------------------------
// Kernel 2: out[n][f] = bias[f]   (also clears the 0xAA poison each call)
// -----------------------------------------------------------------------------
__global__ __launch_bounds__(256) void gcn_init_bias(
    float* __restrict__ out, const float* __restrict__ bias, int total)
{
    int i = blockIdx.x * 256 + threadIdx.x;
    if (i < total) out[i] = bias[i & (F_OUT - 1)];
}

// -----------------------------------------------------------------------------
// Kernel 3: edge scatter. One wave per edge: lane l handles features
// [4l, 4l+4) as a float4 (coalesced 512B per edge). h and out are L2-resident
// on MI455X (51 MB each << 192 MB L2), so gathers + f32 atomics hit L2.
// -----------------------------------------------------------------------------
__global__ __launch_bounds__(256) void gcn_scatter(
    const float* __restrict__ h,
    const int* __restrict__ src, const int* __restrict__ dst,
    const float* __restrict__ vals,
    float* __restrict__ out, int n_edges)
{
    const int lane = threadIdx.x & 31;
    const int e = blockIdx.x * (256 / 32) + (threadIdx.x >> 5);
    if (e >= n_edges) return;

    const int   s = src[e];
    const int   d = dst[e];
    const float v = vals[e];

    const float4 m = *((const float4*)(h + (size_t)s * F_OUT) + lane);
    float* op = out + (size_t)d * F_OUT + lane * 4;
    atomicAdd(op + 0, m.x * v);
    atomicAdd(op + 1, m.y * v);
    atomicAdd(op + 2, m.z * v);
    atomicAdd(op + 3, m.w * v);
}

// -----------------------------------------------------------------------------
// Host-side launch. Inputs (setup_inputs order):
//   0: x [N,256] f32   1: edge_src [E] int   2: edge_dst [E] int
//   3: edge_vals [E] f32   4: kernel [256,128] f32   5: bias [128] f32
// d_ws holds h = [N,128] f32 (51.2 MB).
// -----------------------------------------------------------------------------
extern "C" void kernel_launch(void* const* d_in, const int* in_sizes, int n_in,
                              void* d_out, int out_size, void* d_ws, size_t ws_size,
                              hipStream_t stream)
{
    const float* x    = (const float*)d_in[0];
    const int*   src  = (const int*)d_in[1];
    const int*   dst  = (const int*)d_in[2];
    const float* vals = (const float*)d_in[3];
    const float* w    = (const float*)d_in[4];
    const float* bias = (const float*)d_in[5];
    float* out = (float*)d_out;
    float* h   = (float*)d_ws;

    const int n_nodes = in_sizes[0] / F_IN;
    const int n_edges = in_sizes[1];

    // 1) bias init (clears poison; stream order serializes vs scatter)
    {
        int total  = out_size;
        int blocks = (total + 255) / 256;
        gcn_init_bias<<<blocks, 256, 0, stream>>>(out, bias, total);
    }
    // 2) dense transform h = x @ w via WMMA
    {
        int mtiles = (n_nodes + 15) / 16;
        int blocks = (mtiles + 3) / 4;
        gcn_gemm_wmma<<<blocks, 128, 0, stream>>>(x, w, h, n_nodes);
    }
    // 3) sparse aggregation: one wave per edge
    {
        int blocks = (n_edges + 7) / 8;
        gcn_scatter<<<blocks, 256, 0, stream>>>(h, src, dst, vals, out, n_edges);
    }
}